// LengthRegulator_75685913690796
// MI455X (gfx1250) — compile-verified
//
#include <hip/hip_runtime.h>
#include <stdint.h>

#ifndef __has_builtin
#define __has_builtin(x) 0
#endif

#define GLOBAL_AS __attribute__((address_space(1)))
#define LOCAL_AS  __attribute__((address_space(3)))

#define B_  32
#define T_  512
#define D_  384
#define L_  3584          // T * (DUR_MAX - 1)
#define D4_ 96            // D / 4 floats per float4

typedef float v4f __attribute__((ext_vector_type(4)));
typedef int   v4i __attribute__((ext_vector_type(4)));

// ---------------------------------------------------------------------------
// Kernel 1: per-batch inclusive scan of max(dur,1); 1 block per batch.
// Writes cum[b][t] (int32, max value 4096) and mel_len[b] (int64) at out tail.
// ---------------------------------------------------------------------------
__global__ __launch_bounds__(T_) void lr_scan_kernel(const int* __restrict__ dur,
                                                     int* __restrict__ cum,
                                                     long long* __restrict__ mel_len) {
    __shared__ int s[T_];
    const int b = blockIdx.x;
    const int t = threadIdx.x;

    int v = dur[b * T_ + t];
    if (v < 1) v = 1;
    s[t] = v;
    __syncthreads();

    // Hillis-Steele inclusive scan over 512 elements (9 steps)
    for (int off = 1; off < T_; off <<= 1) {
        int add = (t >= off) ? s[t - off] : 0;
        __syncthreads();
        s[t] += add;
        __syncthreads();
    }

    cum[b * T_ + t] = s[t];
    if (t == T_ - 1) mel_len[b] = (long long)s[t];
}

// ---------------------------------------------------------------------------
// Kernel 2: expansion gather. grid = (L/32, B), 256 threads (8 waves).
// Each wave handles 4 output frames; per frame: binary search in LDS cum,
// then 96x float4 row copy (3 per lane), NT stores to keep x resident in L2.
// ---------------------------------------------------------------------------
__global__ __launch_bounds__(256) void lr_expand_kernel(const float* __restrict__ x,
                                                        const int* __restrict__ cum,
                                                        float* __restrict__ out) {
    __shared__ int s_cum[T_];
    const int b   = blockIdx.y;
    const int tid = threadIdx.x;

#if __has_builtin(__builtin_amdgcn_global_load_async_to_lds_b128) && \
    __has_builtin(__builtin_amdgcn_s_wait_asynccnt)
    // CDNA5 async copy path: 128 lanes each issue one b128 global->LDS DMA.
    if (tid < T_ / 4) {
        GLOBAL_AS v4i* g = (GLOBAL_AS v4i*)(cum + b * T_ + tid * 4);
        LOCAL_AS  v4i* l = (LOCAL_AS  v4i*)(&s_cum[tid * 4]);
        __builtin_amdgcn_global_load_async_to_lds_b128(g, l, 0, 0);
    }
    __builtin_amdgcn_s_wait_asynccnt(0);
#else
    if (tid < T_ / 4) {
        ((int4*)s_cum)[tid] = ((const int4*)(cum + b * T_))[tid];
    }
#endif
    __syncthreads();

    const int mel  = s_cum[T_ - 1];
    const int wave = tid >> 5;   // wave32
    const int lane = tid & 31;

    const v4f* xb = (const v4f*)x + (size_t)b * ((size_t)T_ * D4_);
    v4f*       ob = (v4f*)out     + (size_t)b * ((size_t)L_ * D4_);

    const int row0 = blockIdx.x * 32 + wave * 4;

    for (int r = 0; r < 4; ++r) {
        const int f = row0 + r;

        // searchsorted(cum, f, side='right'): first i with cum[i] > f
        int lo = 0, hi = T_;
        while (lo < hi) {
            int mid = (lo + hi) >> 1;
            if (s_cum[mid] <= f) lo = mid + 1; else hi = mid;
        }
        const int idx = (lo < T_) ? lo : (T_ - 1);

        v4f* dst = ob + (size_t)f * D4_;

        if (f < mel) {                       // wave-uniform branch
            const v4f* src = xb + idx * D4_;
            __builtin_prefetch(src + lane, 0, 3);   // global_prefetch_b8
            #pragma unroll
            for (int c = 0; c < 3; ++c) {
                v4f val = src[lane + c * 32];
                __builtin_nontemporal_store(val, dst + lane + c * 32);
            }
        } else {                             // masked frame -> zeros
            v4f z = {0.f, 0.f, 0.f, 0.f};
            #pragma unroll
            for (int c = 0; c < 3; ++c) {
                __builtin_nontemporal_store(z, dst + lane + c * 32);
            }
        }
    }
}

// ---------------------------------------------------------------------------
extern "C" void kernel_launch(void* const* d_in, const int* in_sizes, int n_in,
                              void* d_out, int out_size, void* d_ws, size_t ws_size,
                              hipStream_t stream) {
    (void)in_sizes; (void)n_in; (void)out_size; (void)ws_size;

    const float* x   = (const float*)d_in[0];
    const int*   dur = (const int*)d_in[1];
    // d_in[2] = target_mel_len scalar (== L_, compile-time constant here)

    float*     out     = (float*)d_out;
    long long* mel_len = (long long*)(out + (size_t)B_ * L_ * D_);  // tuple tail
    int*       cum     = (int*)d_ws;                                // B*T*4 = 64 KB

    lr_scan_kernel<<<dim3(B_), dim3(T_), 0, stream>>>(dur, cum, mel_len);

    dim3 grid(L_ / 32, B_);
    lr_expand_kernel<<<grid, dim3(256), 0, stream>>>(x, cum, out);
}